// ContrastiveLoss_17093969838495
// MI455X (gfx1250) — compile-verified
//
#include <hip/hip_runtime.h>

// ContrastiveLoss collapses algebraically:
//   pos_sum = sum((a-b)^2)                      (= trace(d))
//   sum(d)  = B*(sum(a^2)+sum(b^2)) - 2*dot(colsum(a), colsum(b))
// so no B x B gram matrix is needed -> single streaming pass, HBM-bound
// (128 MB / 23.3 TB/s ~ 5.5 us). Column sums are computed with
// V_WMMA_F32_16X16X4_F32 using an all-ones A matrix (layout-permutation
// invariant since both s1 and s2 see the same permutation).

typedef __attribute__((ext_vector_type(2))) float v2f;
typedef __attribute__((ext_vector_type(8))) float v8f;

#define DDIM 4096
#define BDIM 4096
#define ROWS_PER_GROUP 128   // rows handled by one wave
#define WAVES_PER_BLOCK 8    // 256 threads

__global__ void __launch_bounds__(256)
contrastive_phase1(const float* __restrict__ p1, const float* __restrict__ p2,
                   float* __restrict__ s1, float* __restrict__ s2,
                   double* __restrict__ scal) {
    const int tid  = threadIdx.x;
    const int wave = tid >> 5;
    const int lane = tid & 31;

    const int slab = blockIdx.x * WAVES_PER_BLOCK + wave; // 16-column slab id
    const int c0   = slab * 16;
    const int r0   = blockIdx.y * ROWS_PER_GROUP;

    const int nlo   = lane & 15;  // column within slab
    const int khalf = lane >> 4;  // 0 or 1: which row of the pair

    size_t base = (size_t)(r0 + khalf) * DDIM + (size_t)(c0 + nlo);
    const float* a = p1 + base;
    const float* b = p2 + base;

    v8f acc1 = {};
    v8f acc2 = {};
    v2f onesA;
    onesA[0] = 1.0f; onesA[1] = 1.0f;

    float fS1 = 0.0f, fS2 = 0.0f, fpos = 0.0f;

    // Each iteration covers 4 rows x 16 cols per matrix:
    //   av[0] = row r+khalf, av[1] = row r+khalf+2  (k-order irrelevant: A is ones)
#pragma unroll 4
    for (int it = 0; it < ROWS_PER_GROUP / 4; ++it) {
        v2f av, bv;
        av[0] = a[0];
        av[1] = a[2 * DDIM];
        bv[0] = b[0];
        bv[1] = b[2 * DDIM];

        // D += ones(16x4) x B(4x16): every output row = column sums of the 4 rows
        acc1 = __builtin_amdgcn_wmma_f32_16x16x4_f32(
            false, onesA, false, av, (short)0, acc1, false, false);
        acc2 = __builtin_amdgcn_wmma_f32_16x16x4_f32(
            false, onesA, false, bv, (short)0, acc2, false, false);

        fS1  += av[0] * av[0] + av[1] * av[1];
        fS2  += bv[0] * bv[0] + bv[1] * bv[1];
        float d0 = av[0] - bv[0];
        float d1 = av[1] - bv[1];
        fpos += d0 * d0 + d1 * d1;

        a += 4 * DDIM;
        b += 4 * DDIM;
    }

    // Column-sum output: acc VGPR0 holds one (replicated) row of D.
    // Lanes 0..15 carry the 16 column sums of this slab (any consistent
    // intra-slab permutation is harmless for dot(s1,s2)).
    if (lane < 16) {
        atomicAdd(&s1[c0 + lane], acc1[0]);
        atomicAdd(&s2[c0 + lane], acc2[0]);
    }

    // Block-reduce the three scalar partials, one f64 atomic each per block.
    __shared__ float red[256];
    float vals[3] = {fS1, fS2, fpos};
    for (int k = 0; k < 3; ++k) {
        red[tid] = vals[k];
        __syncthreads();
        for (int s = 128; s > 0; s >>= 1) {
            if (tid < s) red[tid] += red[tid + s];
            __syncthreads();
        }
        if (tid == 0) atomicAdd(&scal[k], (double)red[0]);
        __syncthreads();
    }
}

__global__ void __launch_bounds__(256)
contrastive_phase2(const float* __restrict__ s1, const float* __restrict__ s2,
                   const double* __restrict__ scal, float* __restrict__ out) {
    __shared__ double red[256];
    double dacc = 0.0;
    for (int c = threadIdx.x; c < DDIM; c += 256)
        dacc += (double)s1[c] * (double)s2[c];
    red[threadIdx.x] = dacc;
    __syncthreads();
    for (int s = 128; s > 0; s >>= 1) {
        if (threadIdx.x < s) red[threadIdx.x] += red[threadIdx.x + s];
        __syncthreads();
    }
    if (threadIdx.x == 0) {
        const double S1  = scal[0];
        const double S2  = scal[1];
        const double pos = scal[2];
        const double dot = red[0];
        const double Bd  = (double)BDIM;

        double pos_loss = pos / Bd;                          // mean positive term
        double off_sum  = Bd * (S1 + S2) - 2.0 * dot - pos;  // sum(d) - trace(d)
        double neg_loss = -off_sum / (Bd * (Bd - 1.0));
        out[0] = (float)(pos_loss + neg_loss);
    }
}

extern "C" void kernel_launch(void* const* d_in, const int* in_sizes, int n_in,
                              void* d_out, int out_size, void* d_ws, size_t ws_size,
                              hipStream_t stream) {
    (void)in_sizes; (void)n_in; (void)out_size; (void)ws_size;

    const float* p1 = (const float*)d_in[0];
    const float* p2 = (const float*)d_in[1];
    float* out = (float*)d_out;

    // Workspace layout: [0..4095] s1 (f32), [4096..8191] s2 (f32),
    // then at byte 32768: 3 doubles {S1, S2, pos_sum}.
    float*  s1   = (float*)d_ws;
    float*  s2   = s1 + DDIM;
    double* scal = (double*)((char*)d_ws + 32768);

    hipMemsetAsync(d_ws, 0, 32768 + 3 * sizeof(double), stream);

    // grid.x: 32 blocks x 8 waves = 256 column slabs (4096 cols / 16)
    // grid.y: 32 row groups x 128 rows = 4096 rows; 8192 waves total.
    dim3 grid(DDIM / 16 / WAVES_PER_BLOCK, BDIM / ROWS_PER_GROUP);
    contrastive_phase1<<<grid, 256, 0, stream>>>(p1, p2, s1, s2, scal);
    contrastive_phase2<<<1, 256, 0, stream>>>(s1, s2, scal, out);
}